// LGAB_74947179315535
// MI455X (gfx1250) — compile-verified
//
#include <hip/hip_runtime.h>
#include <math.h>

typedef float v2f __attribute__((ext_vector_type(2)));
typedef float v8f __attribute__((ext_vector_type(8)));

constexpr int Bn = 4, Cdim = 64, C8n = 8, Ccn = 21, Csn = 2;
constexpr int Hn = 64, Wn = 64, HWn = 4096, Hm = 128;
constexpr int PERB = Cdim * HWn;   // 262144

// ---------------- LayerNorm over channel dim of NCHW ----------------
__global__ void __launch_bounds__(256)
k_ln(const float* __restrict__ x, const float* __restrict__ lw,
     const float* __restrict__ lb, float* __restrict__ y) {
    int p = blockIdx.x * 256 + threadIdx.x;     // 0..16383  (b*hw + pix)
    int b = p >> 12, pix = p & 4095;
    const float* xb = x + b * PERB + pix;
    float s = 0.f, s2 = 0.f;
    #pragma unroll
    for (int c = 0; c < Cdim; ++c) { float t = xb[c * HWn]; s += t; s2 += t * t; }
    float mu   = s * (1.0f / Cdim);
    float var  = s2 * (1.0f / Cdim) - mu * mu;
    float rstd = rsqrtf(var + 1e-5f);
    float* yb = y + b * PERB + pix;
    #pragma unroll
    for (int c = 0; c < Cdim; ++c)
        yb[c * HWn] = (xb[c * HWn] - mu) * rstd * lw[c] + lb[c];
}

// ---------------- 1x1 conv ----------------
__global__ void __launch_bounds__(256)
k_conv1x1(const float* __restrict__ in, const float* __restrict__ w,
          const float* __restrict__ bias, float* __restrict__ out, int Oc) {
    int idx = blockIdx.x * 256 + threadIdx.x;   // B*Oc*HW
    int pix = idx & 4095; int t = idx >> 12; int o = t % Oc; int b = t / Oc;
    const float* ib = in + b * PERB + pix;
    const float* wr = w + o * Cdim;
    float s = bias[o];
    #pragma unroll
    for (int c = 0; c < Cdim; ++c) s += wr[c] * ib[c * HWn];
    out[idx] = s;
}

// ---------------- depthwise 3x3, pad=1 ----------------
__global__ void __launch_bounds__(256)
k_dw3x3(const float* __restrict__ in, const float* __restrict__ w,
        const float* __restrict__ bias, float* __restrict__ out, int Cch) {
    int idx = blockIdx.x * 256 + threadIdx.x;   // B*Cch*HW
    int pix = idx & 4095; int bc = idx >> 12;
    int c = bc % Cch;
    int y = pix >> 6, x = pix & 63;
    const float* plane = in + bc * HWn;
    const float* wr = w + c * 9;
    float s = bias[c];
    #pragma unroll
    for (int dy = 0; dy < 3; ++dy) {
        int iy = y + dy - 1;
        if ((unsigned)iy >= (unsigned)Hn) continue;
        #pragma unroll
        for (int dx = 0; dx < 3; ++dx) {
            int ix = x + dx - 1;
            if ((unsigned)ix >= (unsigned)Wn) continue;
            s += wr[dy * 3 + dx] * plane[iy * Wn + ix];
        }
    }
    out[idx] = s;
}

// ---------------- flash attention ----------------
// One wave per 16-query tile; 8 waves/block share K/V key tiles staged in LDS
// by double-buffered global_load_async_to_lds (ASYNCcnt), overlapping DMA of
// tile t+1 with WMMA compute on tile t.
// q,k: (B, 8, HW)   v: (B, 64, HW)   out: (B, 64, HW)
__global__ void __launch_bounds__(256)
k_attn(const float* __restrict__ q, const float* __restrict__ k,
       const float* __restrict__ v, float* __restrict__ out) {
    __shared__ float Kbuf[2][8][16];            // 1 KB   : K tile double buffer
    __shared__ float Vbuf[2][64][16];           // 8 KB   : V tile double buffer
    __shared__ float Pst[8][16][16];            // 8 KB   : per-wave P staging

    const int tid  = threadIdx.x;
    const int wave = tid >> 5;
    const int lane = tid & 31;
    const int gw   = blockIdx.x * 8 + wave;     // 1024 waves total
    const int b    = gw >> 8;                   // 256 query tiles per batch
    const int qt   = gw & 255;
    const int n    = lane & 15;
    const int hi   = lane >> 4;

    const float* qb = q + b * (C8n * HWn);
    const float* kb = k + b * (C8n * HWn);
    const float* vb = v + b * PERB;

    // async-staging address split: all 256 threads move 16B of V, first 128 move 4B of K
    const int vch = tid >> 2, vk4 = (tid & 3) * 4;
    const int kdd = tid >> 4, kkk = tid & 15;

    auto issue_tile = [&](int jn, int bufi) {
        const float* gsrc = vb + vch * HWn + jn + vk4;
        unsigned ldv = (unsigned)(size_t)(&Vbuf[bufi][vch][vk4]);
        asm volatile("global_load_async_to_lds_b128 %0, %1, off"
                     :: "v"(ldv), "v"(gsrc) : "memory");
        if (tid < 128) {
            const float* gk = kb + kdd * HWn + jn + kkk;
            unsigned ldk = (unsigned)(size_t)(&Kbuf[bufi][kdd][kkk]);
            asm volatile("global_load_async_to_lds_b32 %0, %1, off"
                         :: "v"(ldk), "v"(gk) : "memory");
        }
    };

    // A fragment (Q tile, 16x8): lane row m = qt*16+n; VGPR j holds d = kc*4 + 2*hi + j
    const int m = qt * 16 + n;
    v2f Aq[2];
    #pragma unroll
    for (int kc = 0; kc < 2; ++kc) {
        int d0 = kc * 4 + 2 * hi;
        Aq[kc].x = qb[(d0 + 0) * HWn + m];
        Aq[kc].y = qb[(d0 + 1) * HWn + m];
    }

    const v8f vzero = {0.f,0.f,0.f,0.f,0.f,0.f,0.f,0.f};
    v8f acc[4];
    #pragma unroll
    for (int f = 0; f < 4; ++f) acc[f] = vzero;
    float mrun[8], lrun[8];
    #pragma unroll
    for (int r = 0; r < 8; ++r) { mrun[r] = -1e30f; lrun[r] = 0.f; }

    issue_tile(0, 0);                           // prologue: DMA tile 0 -> buf 0

    for (int kt = 0; kt < 256; ++kt) {
        const int cur = kt & 1, nxt = cur ^ 1;
        // my DMA of tile kt complete; barrier publishes it to all waves and
        // guarantees everyone is done reading buf[nxt] (tile kt-1)
        asm volatile("s_wait_asynccnt 0" ::: "memory");
        __syncthreads();
        if (kt + 1 < 256) issue_tile((kt + 1) * 16, nxt);   // overlap DMA with compute

        // S = Q . K^T  (two f32 WMMA, K=4 each, covering d=0..7)
        v8f s = vzero;
        #pragma unroll
        for (int kc = 0; kc < 2; ++kc) {
            int d0 = kc * 4 + 2 * hi;
            v2f Bk;
            Bk.x = Kbuf[cur][d0 + 0][n];
            Bk.y = Kbuf[cur][d0 + 1][n];
            s = __builtin_amdgcn_wmma_f32_16x16x4_f32(false, Aq[kc], false, Bk,
                                                      (short)0, s, false, false);
        }

        // online softmax: row r (+8*hi) is spread over the 16 lanes of this half-wave
        #pragma unroll
        for (int r = 0; r < 8; ++r) {
            float val  = s[r];
            float rmax = val;
            #pragma unroll
            for (int off = 1; off < 16; off <<= 1)
                rmax = fmaxf(rmax, __shfl_xor(rmax, off, 32));
            float mn   = fmaxf(mrun[r], rmax);
            float pval = __expf(val - mn);
            float rs   = pval;
            #pragma unroll
            for (int off = 1; off < 16; off <<= 1)
                rs += __shfl_xor(rs, off, 32);
            float corr = __expf(mrun[r] - mn);
            lrun[r] = lrun[r] * corr + rs;
            mrun[r] = mn;
            #pragma unroll
            for (int f = 0; f < 4; ++f) acc[f][r] *= corr;
            Pst[wave][r + 8 * hi][n] = pval;
        }
        asm volatile("s_wait_dscnt 0" ::: "memory");

        // re-fetch P in A-fragment layout (wave-private LDS, DS in-order per wave)
        v2f Ap[4];
        #pragma unroll
        for (int kc = 0; kc < 4; ++kc)
            Ap[kc] = *(const v2f*)(&Pst[wave][n][kc * 4 + 2 * hi]);

        // acc += P(16x16) x V^T(16x64): 4 channel tiles x 4 K-chunks of f32 WMMA
        #pragma unroll
        for (int ct = 0; ct < 4; ++ct) {
            const float* vrow = &Vbuf[cur][ct * 16 + n][0];
            #pragma unroll
            for (int kc = 0; kc < 4; ++kc) {
                v2f Bv = *(const v2f*)(vrow + kc * 4 + 2 * hi);
                acc[ct] = __builtin_amdgcn_wmma_f32_16x16x4_f32(false, Ap[kc], false, Bv,
                                                                (short)0, acc[ct], false, false);
            }
        }
    }

    // normalize and write: D layout -> lane col n = channel, VGPR r -> query row r+8*hi
    float* ob = out + b * PERB;
    #pragma unroll
    for (int r = 0; r < 8; ++r) {
        float inv = 1.0f / lrun[r];
        int qi = qt * 16 + r + 8 * hi;
        #pragma unroll
        for (int ct = 0; ct < 4; ++ct)
            ob[(ct * 16 + n) * HWn + qi] = acc[ct][r] * inv;
    }
}

// ---------------- proj 1x1 + gamma*out + xn ----------------
__global__ void __launch_bounds__(256)
k_proj(const float* __restrict__ sa, const float* __restrict__ w,
       const float* __restrict__ bias, const float* __restrict__ gamma,
       const float* __restrict__ xn, float* __restrict__ out) {
    int idx = blockIdx.x * 256 + threadIdx.x;   // B*64*HW
    int pix = idx & 4095; int t = idx >> 12; int o = t & 63; int b = t >> 6;
    const float* ib = sa + b * PERB + pix;
    const float* wr = w + o * Cdim;
    float s = bias[o];
    #pragma unroll
    for (int c = 0; c < Cdim; ++c) s += wr[c] * ib[c * HWn];
    out[idx] = gamma[0] * s + xn[idx];
}

// ---------------- dense 3x3 conv, pad=1, optional exact GELU ----------------
__global__ void __launch_bounds__(256)
k_conv3x3(const float* __restrict__ in, const float* __restrict__ w,
          const float* __restrict__ bias, float* __restrict__ out,
          int Ic, int Oc, int apply_gelu) {
    int idx = blockIdx.x * 256 + threadIdx.x;   // B*Oc*HW
    int pix = idx & 4095; int t = idx >> 12; int o = t % Oc; int b = t / Oc;
    int y = pix >> 6, x = pix & 63;
    const float* ib = in + b * Ic * HWn;
    const float* wr = w + o * Ic * 9;
    float s = bias[o];
    for (int c = 0; c < Ic; ++c) {
        const float* plane = ib + c * HWn;
        const float* wc = wr + c * 9;
        #pragma unroll
        for (int dy = 0; dy < 3; ++dy) {
            int iy = y + dy - 1;
            if ((unsigned)iy >= (unsigned)Hn) continue;
            #pragma unroll
            for (int dx = 0; dx < 3; ++dx) {
                int ix = x + dx - 1;
                if ((unsigned)ix >= (unsigned)Wn) continue;
                s += wc[dy * 3 + dx] * plane[iy * Wn + ix];
            }
        }
    }
    if (apply_gelu) s = 0.5f * s * (1.0f + erff(s * 0.70710678118654752f));
    out[idx] = s;
}

// ---------------- global average pool: one block per (b,c) ----------------
__global__ void __launch_bounds__(256)
k_pool(const float* __restrict__ y, float* __restrict__ p) {
    __shared__ float red[256];
    int bc = blockIdx.x;
    const float* plane = y + bc * HWn;
    float s = 0.f;
    for (int i = threadIdx.x; i < HWn; i += 256) s += plane[i];
    red[threadIdx.x] = s;
    __syncthreads();
    for (int st = 128; st > 0; st >>= 1) {
        if (threadIdx.x < st) red[threadIdx.x] += red[threadIdx.x + st];
        __syncthreads();
    }
    if (threadIdx.x == 0) p[bc] = red[0] * (1.0f / HWn);
}

// ---------------- channel attention squeeze/excite ----------------
__global__ void __launch_bounds__(256)
k_ca(const float* __restrict__ pool,
     const float* __restrict__ w1, const float* __restrict__ b1,
     const float* __restrict__ w2, const float* __restrict__ b2,
     float* __restrict__ scale) {
    __shared__ float p2[Bn][Csn];
    int tid = threadIdx.x;
    if (tid < Bn * Csn) {
        int b = tid / Csn, cs = tid % Csn;
        float s = b1[cs];
        for (int c = 0; c < Cdim; ++c) s += w1[cs * Cdim + c] * pool[b * Cdim + c];
        p2[b][cs] = fmaxf(s, 0.f);
    }
    __syncthreads();
    int b = tid >> 6, c = tid & 63;
    float s = b2[c];
    #pragma unroll
    for (int j = 0; j < Csn; ++j) s += w2[c * Csn + j] * p2[b][j];
    scale[tid] = 1.0f / (1.0f + __expf(-s));
}

// ---------------- feat_sum = sa_feat + y*ca_scale + shortcut ----------------
__global__ void __launch_bounds__(256)
k_featsum(const float* __restrict__ sa_feat, const float* __restrict__ ycab,
          const float* __restrict__ scale, const float* __restrict__ x,
          float* __restrict__ out) {
    int idx = blockIdx.x * 256 + threadIdx.x;   // B*C*HW
    int bc = idx >> 12;
    out[idx] = sa_feat[idx] + ycab[idx] * scale[bc] + x[idx];
}

// ---------------- fc1 + exact GELU: t1[p*128+h] ----------------
__global__ void __launch_bounds__(256)
k_fc1(const float* __restrict__ fs1, const float* __restrict__ w,
      const float* __restrict__ bias, float* __restrict__ t1) {
    int idx = blockIdx.x * 256 + threadIdx.x;   // 16384*128
    int h = idx & 127; int p = idx >> 7;
    int b = p >> 12, pix = p & 4095;
    const float* ib = fs1 + b * PERB + pix;
    const float* wr = w + h * Cdim;
    float s = bias[h];
    #pragma unroll
    for (int c = 0; c < Cdim; ++c) s += wr[c] * ib[c * HWn];
    t1[idx] = 0.5f * s * (1.0f + erff(s * 0.70710678118654752f));
}

// ---------------- fc2 + residual, with the reference's raw-view reshape ----------------
__global__ void __launch_bounds__(256)
k_fc2(const float* __restrict__ t1, const float* __restrict__ w,
      const float* __restrict__ bias, const float* __restrict__ feat,
      float* __restrict__ out) {
    int idx = blockIdx.x * 256 + threadIdx.x;   // B*C*HW, flat == (b, hw, c) raw view
    int c = idx & 63; int pg = idx >> 6;        // global pixel id 0..16383
    const float* tr = t1 + pg * Hm;
    const float* wr = w + c * Hm;
    float s = bias[c];
    #pragma unroll
    for (int h = 0; h < Hm; ++h) s += wr[h] * tr[h];
    out[idx] = s + feat[idx];
}

extern "C" void kernel_launch(void* const* d_in, const int* in_sizes, int n_in,
                              void* d_out, int out_size, void* d_ws, size_t ws_size,
                              hipStream_t stream) {
    (void)in_sizes; (void)n_in; (void)out_size; (void)ws_size;
    const float* x      = (const float*)d_in[0];
    const float* ln_w   = (const float*)d_in[1];
    const float* ln_b   = (const float*)d_in[2];
    const float* q_w    = (const float*)d_in[3];
    const float* q_b    = (const float*)d_in[4];
    const float* k_w    = (const float*)d_in[5];
    const float* k_b    = (const float*)d_in[6];
    const float* v_w    = (const float*)d_in[7];
    const float* v_b    = (const float*)d_in[8];
    const float* qkdw_w = (const float*)d_in[9];
    const float* qkdw_b = (const float*)d_in[10];
    const float* vdw_w  = (const float*)d_in[11];
    const float* vdw_b  = (const float*)d_in[12];
    const float* proj_w = (const float*)d_in[13];
    const float* proj_b = (const float*)d_in[14];
    const float* gamma  = (const float*)d_in[15];
    const float* cab_w1 = (const float*)d_in[16];
    const float* cab_b1 = (const float*)d_in[17];
    const float* cab_w2 = (const float*)d_in[18];
    const float* cab_b2 = (const float*)d_in[19];
    const float* ca_w1  = (const float*)d_in[20];
    const float* ca_b1  = (const float*)d_in[21];
    const float* ca_w2  = (const float*)d_in[22];
    const float* ca_b2  = (const float*)d_in[23];
    const float* fc1_w  = (const float*)d_in[24];
    const float* fc1_b  = (const float*)d_in[25];
    const float* fc2_w  = (const float*)d_in[26];
    const float* fc2_b  = (const float*)d_in[27];
    float* out = (float*)d_out;

    // workspace layout (floats): total ~11.4M floats (~45.4 MB)
    float* ws      = (float*)d_ws;
    float* xn      = ws;                 // 1048576
    float* q0      = xn + 1048576;       // 131072
    float* k0      = q0 + 131072;        // 131072
    float* qd      = k0 + 131072;        // 131072
    float* kd      = qd + 131072;        // 131072
    float* v0      = kd + 131072;        // 1048576
    float* vd      = v0 + 1048576;       // 1048576
    float* sa_out  = vd + 1048576;       // 1048576
    float* sa_feat = sa_out + 1048576;   // 1048576
    float* ycab1   = sa_feat + 1048576;  // 344064  (B,21,HW)
    float* ycab    = ycab1 + 344064;     // 1048576
    float* feat    = ycab + 1048576;     // 1048576
    float* fs1     = feat + 1048576;     // 1048576
    float* t1      = fs1 + 1048576;      // 2097152 (B*HW, 128)
    float* pool    = t1 + 2097152;       // 256
    float* cascale = pool + 256;         // 256

    // 1) xn = LayerNorm_c(x)
    k_ln<<<64, 256, 0, stream>>>(x, ln_w, ln_b, xn);
    // 2) 1x1 convs -> q0/k0/v0
    k_conv1x1<<<512, 256, 0, stream>>>(xn, q_w, q_b, q0, C8n);
    k_conv1x1<<<512, 256, 0, stream>>>(xn, k_w, k_b, k0, C8n);
    k_conv1x1<<<4096, 256, 0, stream>>>(xn, v_w, v_b, v0, Cdim);
    // 3) depthwise 3x3
    k_dw3x3<<<512, 256, 0, stream>>>(q0, qkdw_w, qkdw_b, qd, C8n);
    k_dw3x3<<<512, 256, 0, stream>>>(k0, qkdw_w, qkdw_b, kd, C8n);
    k_dw3x3<<<4096, 256, 0, stream>>>(v0, vdw_w, vdw_b, vd, Cdim);
    // 4) flash attention (WMMA f32 16x16x4 + async-to-LDS double buffering)
    k_attn<<<128, 256, 0, stream>>>(qd, kd, vd, sa_out);
    // 5) proj + gamma + xn
    k_proj<<<4096, 256, 0, stream>>>(sa_out, proj_w, proj_b, gamma, xn, sa_feat);
    // 6) CAB: conv3x3 (gelu) -> conv3x3
    k_conv3x3<<<1344, 256, 0, stream>>>(xn, cab_w1, cab_b1, ycab1, Cdim, Ccn, 1);
    k_conv3x3<<<4096, 256, 0, stream>>>(ycab1, cab_w2, cab_b2, ycab, Ccn, Cdim, 0);
    // 7) channel attention
    k_pool<<<256, 256, 0, stream>>>(ycab, pool);
    k_ca<<<1, 256, 0, stream>>>(pool, ca_w1, ca_b1, ca_w2, ca_b2, cascale);
    // 8) feat_sum = sa_feat + ycab*scale + x
    k_featsum<<<4096, 256, 0, stream>>>(sa_feat, ycab, cascale, x, feat);
    // 9) MLP: LN -> fc1+gelu -> fc2 + residual (raw-view reshape)
    k_ln<<<64, 256, 0, stream>>>(feat, ln_w, ln_b, fs1);
    k_fc1<<<8192, 256, 0, stream>>>(fs1, fc1_w, fc1_b, t1);
    k_fc2<<<4096, 256, 0, stream>>>(t1, fc2_w, fc2_b, feat, out);
}